// YOLO6DLoss_12584254177657
// MI455X (gfx1250) — compile-verified
//
#include <hip/hip_runtime.h>
#include <hip/hip_bf16.h>

typedef __attribute__((ext_vector_type(2))) float v2f;
typedef __attribute__((ext_vector_type(8))) float v8f;

#define B_N 262144
#define NP_N 64

// ---------------------------------------------------------------------------
// Wave32 full-lane sum using V_WMMA_F32_16X16X4_F32 against an all-ones B.
// A layout (CDNA5 ISA, 32-bit A 16x4): VGPR0 = K=0 (lanes 0-15) / K=2
// (lanes 16-31), VGPR1 = K=1 / K=3. With a1=0 and B=ones, row sum
// s[m] = x[m] + x[m+16], replicated across all 16 columns of D.
// D layout: VGPR r holds M=r (lanes 0-15) and M=r+8 (lanes 16-31), so
// sum(d[0..7]) = sum(s[0..7]) in lanes 0-15 and sum(s[8..15]) in lanes 16-31.
// One shfl_xor(16) then yields the full 32-lane sum in every lane.
// Requires EXEC all-ones: only call from fully-converged code.
// ---------------------------------------------------------------------------
__device__ __forceinline__ float wave_sum_wmma(float x) {
  v2f a;    a[0] = x;    a[1] = 0.0f;
  v2f ones; ones[0] = 1.0f; ones[1] = 1.0f;
  v8f c = {};
  v8f d = __builtin_amdgcn_wmma_f32_16x16x4_f32(
      /*neg_a=*/false, a, /*neg_b=*/false, ones,
      /*c_mod=*/(short)0, c, /*reuse_a=*/false, /*reuse_b=*/false);
  float t = d[0] + d[1] + d[2] + d[3] + d[4] + d[5] + d[6] + d[7];
  t += __shfl_xor(t, 16, 32);
  return t;
}

// ---------------------------------------------------------------------------
// Special orthogonal Procrustes (matches jnp svd_orthogonalize) + geodesic
// angle vs. gt. Jacobi eigen of M^T M (4 cyclic sweeps is plenty for 3x3),
// eigenvalues sorted descending, U columns from M*v_j normalized, last column
// sign-flipped by sign(det M).
// ---------------------------------------------------------------------------
__device__ __forceinline__ float geo_angle(const float* m, const float* g) {
  float A[3][3];
#pragma unroll
  for (int i = 0; i < 3; i++)
#pragma unroll
    for (int j = 0; j < 3; j++) {
      float s = 0.0f;
#pragma unroll
      for (int k = 0; k < 3; k++) s = fmaf(m[k * 3 + i], m[k * 3 + j], s);
      A[i][j] = s;
    }
  float V[3][3] = {{1.f, 0.f, 0.f}, {0.f, 1.f, 0.f}, {0.f, 0.f, 1.f}};

#pragma unroll
  for (int sweep = 0; sweep < 4; sweep++) {
#pragma unroll
    for (int pair = 0; pair < 3; pair++) {
      const int p = (pair == 2) ? 1 : 0;
      const int q = (pair == 0) ? 1 : 2;
      float apq = A[p][q];
      if (fabsf(apq) > 1e-20f) {
        float tau = (A[q][q] - A[p][p]) / (2.0f * apq);
        float t = copysignf(1.0f, tau) / (fabsf(tau) + sqrtf(fmaf(tau, tau, 1.0f)));
        float c = rsqrtf(fmaf(t, t, 1.0f));
        float s = t * c;
#pragma unroll
        for (int k = 0; k < 3; k++) {           // column transform: A*J
          float akp = A[k][p], akq = A[k][q];
          A[k][p] = c * akp - s * akq;
          A[k][q] = s * akp + c * akq;
        }
#pragma unroll
        for (int k = 0; k < 3; k++) {           // row transform: J^T*A
          float apk = A[p][k], aqk = A[q][k];
          A[p][k] = c * apk - s * aqk;
          A[q][k] = s * apk + c * aqk;
        }
#pragma unroll
        for (int k = 0; k < 3; k++) {           // V = V*J
          float vkp = V[k][p], vkq = V[k][q];
          V[k][p] = c * vkp - s * vkq;
          V[k][q] = s * vkp + c * vkq;
        }
      }
    }
  }

  float l0 = A[0][0], l1 = A[1][1], l2 = A[2][2];
  if (l0 < l1) {
    float t = l0; l0 = l1; l1 = t;
#pragma unroll
    for (int k = 0; k < 3; k++) { float v = V[k][0]; V[k][0] = V[k][1]; V[k][1] = v; }
  }
  if (l0 < l2) {
    float t = l0; l0 = l2; l2 = t;
#pragma unroll
    for (int k = 0; k < 3; k++) { float v = V[k][0]; V[k][0] = V[k][2]; V[k][2] = v; }
  }
  if (l1 < l2) {
    float t = l1; l1 = l2; l2 = t;
#pragma unroll
    for (int k = 0; k < 3; k++) { float v = V[k][1]; V[k][1] = V[k][2]; V[k][2] = v; }
  }

  float U0[3], U1[3], U2[3];
#pragma unroll
  for (int i = 0; i < 3; i++) {
    U0[i] = fmaf(m[i * 3 + 0], V[0][0], fmaf(m[i * 3 + 1], V[1][0], m[i * 3 + 2] * V[2][0]));
    U1[i] = fmaf(m[i * 3 + 0], V[0][1], fmaf(m[i * 3 + 1], V[1][1], m[i * 3 + 2] * V[2][1]));
    U2[i] = fmaf(m[i * 3 + 0], V[0][2], fmaf(m[i * 3 + 1], V[1][2], m[i * 3 + 2] * V[2][2]));
  }
  float n0 = rsqrtf(fmaf(U0[0], U0[0], fmaf(U0[1], U0[1], U0[2] * U0[2])) + 1e-30f);
  float n1 = rsqrtf(fmaf(U1[0], U1[0], fmaf(U1[1], U1[1], U1[2] * U1[2])) + 1e-30f);
  float n2 = rsqrtf(fmaf(U2[0], U2[0], fmaf(U2[1], U2[1], U2[2] * U2[2])) + 1e-30f);
#pragma unroll
  for (int i = 0; i < 3; i++) { U0[i] *= n0; U1[i] *= n1; U2[i] *= n2; }

  float detM = m[0] * (m[4] * m[8] - m[5] * m[7])
             - m[1] * (m[3] * m[8] - m[5] * m[6])
             + m[2] * (m[3] * m[7] - m[4] * m[6]);
  float sgn = (detM < 0.0f) ? -1.0f : 1.0f;

  float tr = 0.0f;
#pragma unroll
  for (int i = 0; i < 3; i++)
#pragma unroll
    for (int j = 0; j < 3; j++) {
      float r = fmaf(U0[i], V[j][0], fmaf(U1[i], V[j][1], sgn * U2[i] * V[j][2]));
      tr = fmaf(r, g[i * 3 + j], tr);
    }
  float cs = fminf(fmaxf((tr - 1.0f) * 0.5f, -1.0f + 1e-7f), 1.0f - 1e-7f);
  return acosf(cs);
}

// ---------------------------------------------------------------------------
// Pass 1: fused streaming partial reduction of all 5 loss components.
// ---------------------------------------------------------------------------
__global__ __launch_bounds__(256) void yolo6d_partial(
    const float* __restrict__ rot_pred, const float* __restrict__ rot_gt,
    const float* __restrict__ depth_pred, const float* __restrict__ depth_gt,
    const float4* __restrict__ kpt_pred4, const float4* __restrict__ kpt_gt4,
    const float2* __restrict__ vis2, const float4* __restrict__ box_pred4,
    const float4* __restrict__ box_gt4, float* __restrict__ partials,
    int nthreads) {
  const int tid = blockIdx.x * blockDim.x + threadIdx.x;

  float acc_geo = 0.0f, acc_sl1 = 0.0f, acc_kw = 0.0f, acc_vis = 0.0f, acc_bb = 0.0f;

  // L_R: geodesic after SVD orthogonalization
  for (int i = tid; i < B_N; i += nthreads) {
    float m[9], g[9];
#pragma unroll
    for (int k = 0; k < 9; k++) { m[k] = rot_pred[i * 9 + k]; g[k] = rot_gt[i * 9 + k]; }
    acc_geo += geo_angle(m, g);
  }

  // L_t: smooth L1 on depth
  for (int i = tid; i < B_N; i += nthreads) {
    float a = fabsf(depth_pred[i] - depth_gt[i]);
    acc_sl1 += (a < 1.0f) ? 0.5f * a * a : (a - 0.5f);
  }

  // L_kp: visibility-masked squared distances (dominant HBM stream)
  const int KP4 = B_N * NP_N / 2;  // two keypoints per float4
  for (int i = tid; i < KP4; i += nthreads) {
    __builtin_prefetch((const void*)(kpt_pred4 + i + nthreads), 0, 0);
    __builtin_prefetch((const void*)(kpt_gt4 + i + nthreads), 0, 0);
    float4 p = kpt_pred4[i];
    float4 q = kpt_gt4[i];
    float2 v = vis2[i];
    float dx0 = p.x - q.x, dy0 = p.y - q.y;
    float dx1 = p.z - q.z, dy1 = p.w - q.w;
    acc_kw = fmaf(v.x, fmaf(dx0, dx0, dy0 * dy0), acc_kw);
    acc_kw = fmaf(v.y, fmaf(dx1, dx1, dy1 * dy1), acc_kw);
    acc_vis += v.x + v.y;
  }

  // L_bb: L1 bbox
  for (int i = tid; i < B_N; i += nthreads) {
    float4 p = box_pred4[i];
    float4 q = box_gt4[i];
    acc_bb += fabsf(p.x - q.x) + fabsf(p.y - q.y) + fabsf(p.z - q.z) + fabsf(p.w - q.w);
  }

  // Fully-converged here: WMMA-based wave32 reductions (EXEC all ones).
  float r0 = wave_sum_wmma(acc_geo);
  float r1 = wave_sum_wmma(acc_sl1);
  float r2 = wave_sum_wmma(acc_kw);
  float r3 = wave_sum_wmma(acc_vis);
  float r4 = wave_sum_wmma(acc_bb);

  __shared__ float smem[8][5];
  const int wave = threadIdx.x >> 5;
  const int lane = threadIdx.x & 31;
  if (lane == 0) {
    smem[wave][0] = r0; smem[wave][1] = r1; smem[wave][2] = r2;
    smem[wave][3] = r3; smem[wave][4] = r4;
  }
  __syncthreads();
  if (threadIdx.x == 0) {
    float s0 = 0, s1 = 0, s2 = 0, s3 = 0, s4 = 0;
#pragma unroll
    for (int w = 0; w < 8; w++) {
      s0 += smem[w][0]; s1 += smem[w][1]; s2 += smem[w][2];
      s3 += smem[w][3]; s4 += smem[w][4];
    }
    float* o = partials + (size_t)blockIdx.x * 5;
    o[0] = s0; o[1] = s1; o[2] = s2; o[3] = s3; o[4] = s4;
  }
}

// ---------------------------------------------------------------------------
// Pass 2: deterministic finalize (fixed reduction tree, no atomics).
// Waves 0..4 each own one component.
// ---------------------------------------------------------------------------
__global__ __launch_bounds__(160) void yolo6d_finalize(
    const float* __restrict__ partials, int nb, float* __restrict__ out) {
  __shared__ float comp[5];
  const int wave = threadIdx.x >> 5;
  const int lane = threadIdx.x & 31;
  float s = 0.0f;
  for (int i = lane; i < nb; i += 32) s += partials[(size_t)i * 5 + wave];
#pragma unroll
  for (int off = 16; off > 0; off >>= 1) s += __shfl_xor(s, off, 32);
  if (lane == 0) comp[wave] = s;
  __syncthreads();
  if (threadIdx.x == 0) {
    const float Bf = (float)B_N;
    float l_r = comp[0] / Bf;
    float l_t = comp[1] / Bf;
    float denom = comp[3];
    float l_kp = (denom > 0.0f) ? (comp[2] / fmaxf(denom, 1e-9f)) : 0.0f;
    float l_bb = comp[4] / (4.0f * Bf);
    out[0] = l_r + l_t + l_kp + l_bb;
    out[1] = l_r;
    out[2] = l_t;
    out[3] = l_kp;
    out[4] = l_bb;
  }
}

extern "C" void kernel_launch(void* const* d_in, const int* in_sizes, int n_in,
                              void* d_out, int out_size, void* d_ws, size_t ws_size,
                              hipStream_t stream) {
  const float*  rot_pred   = (const float*)d_in[0];
  const float*  rot_gt     = (const float*)d_in[1];
  const float*  depth_pred = (const float*)d_in[2];
  const float*  depth_gt   = (const float*)d_in[3];
  const float4* kpt_pred4  = (const float4*)d_in[4];
  const float4* kpt_gt4    = (const float4*)d_in[5];
  const float2* vis2       = (const float2*)d_in[6];
  const float4* box_pred4  = (const float4*)d_in[7];
  const float4* box_gt4    = (const float4*)d_in[8];

  int nb = 1024;
  size_t need = (size_t)nb * 5 * sizeof(float);
  if (ws_size < need) {
    nb = (int)(ws_size / (5 * sizeof(float)));
    if (nb < 1) nb = 1;
  }
  float* partials = (float*)d_ws;
  int nthreads = nb * 256;

  yolo6d_partial<<<nb, 256, 0, stream>>>(rot_pred, rot_gt, depth_pred, depth_gt,
                                         kpt_pred4, kpt_gt4, vis2, box_pred4,
                                         box_gt4, partials, nthreads);
  yolo6d_finalize<<<1, 160, 0, stream>>>(partials, nb, (float*)d_out);
}